// PointDSC_5669356835845
// MI455X (gfx1250) — compile-verified
//
#include <hip/hip_runtime.h>
#include <hip/hip_bf16.h>
#include <math.h>
#include <stdint.h>

#define N_PTS 2048
#define CH    128
#define BS    2
#define NSEED 204          // int(2048 * 0.1)
#define KNNK  40
#define SIG_D2_INV 100.0f  // 1 / SIGMA_D^2, SIGMA_D = 0.1
#define BN_EPS 1e-5f

typedef __attribute__((ext_vector_type(2))) float v2f;
typedef __attribute__((ext_vector_type(8))) float v8f;

// ---------------------------------------------------------------------------
// CDNA5 async global->LDS copy (ASYNCcnt-tracked, no VGPR round trip).
// lds_off: LDS byte address (low 32 bits of a flat shared pointer).
__device__ __forceinline__ void async_ld_b128(unsigned lds_off, const float* g)
{
    asm volatile("global_load_async_to_lds_b128 %0, %1, off"
                 :: "v"(lds_off), "v"(g) : "memory");
}
__device__ __forceinline__ void wait_asynccnt0()
{
    asm volatile("s_wait_asynccnt 0x0" ::: "memory");
}

// ---------------------------------------------------------------------------
// WMMA f32 GEMM (NT):  C[M,N] = scale * (A x B^T) + bias[n] + addsrc (opt relu)
//   A: f32 [M,K] row-major,  B: f32 [N,K] row-major.
// Block: 256 threads (8 waves), 64x64 tile; wave = 16(M) x 32(N) (2 accums).
// K-panels of 32 double-buffered in LDS via async global->LDS loads.
// Requires M%64==0, N%64==0, K%32==0 (true for all uses here).
// ---------------------------------------------------------------------------
__global__ __launch_bounds__(256) void gemm_wmma_f32(
    const float* __restrict__ A, const float* __restrict__ B,
    float* __restrict__ C, const float* __restrict__ bias,
    const float* __restrict__ addsrc,
    int M, int N, int K, long sA, long sB, long sC, float scale, int relu)
{
    const int bz = blockIdx.z;
    A += (long)bz * sA;
    B += (long)bz * sB;
    C += (long)bz * sC;
    if (addsrc) addsrc += (long)bz * sC;

    const int m0 = blockIdx.y * 64;
    const int n0 = blockIdx.x * 64;

    __shared__ __align__(16) float As[2][64 * 32];
    __shared__ __align__(16) float Bs[2][64 * 32];

    const int t    = threadIdx.x;
    const int lane = t & 31;
    const int wv   = t >> 5;      // wave 0..7
    const int wm   = wv & 3;      // 4 waves along M (16 each)
    const int wn   = wv >> 2;     // 2 waves along N (32 each)
    const int lrow = lane & 15;
    const int lhi  = lane >> 4;   // 0/1

    // per-thread staging slot: row r (0..63), k-chunk c0 (0,8,16,24)
    const int r  = t >> 2;
    const int c0 = (t & 3) * 8;
    const unsigned eoff  = (unsigned)(r * 32 + c0) * 4u;
    const unsigned abase = (unsigned)(uintptr_t)(&As[0][0]);
    const unsigned bbase = (unsigned)(uintptr_t)(&Bs[0][0]);
    const float* gA = A + (long)(m0 + r) * K + c0;
    const float* gB = B + (long)(n0 + r) * K + c0;

    v8f acc0 = {};
    v8f acc1 = {};

    // prologue: async-stage K-panel 0 into buffer 0
    async_ld_b128(abase + eoff,      gA);
    async_ld_b128(abase + eoff + 16, gA + 4);
    async_ld_b128(bbase + eoff,      gB);
    async_ld_b128(bbase + eoff + 16, gB + 4);

    int buf = 0;
    for (int kk = 0; kk < K; kk += 32) {
        wait_asynccnt0();        // own async loads into As[buf]/Bs[buf] done
        __syncthreads();         // everyone's loads done; prev reads of buf^1 done
        if (kk + 32 < K) {       // async-stage next panel into the other buffer
            const unsigned nb = (unsigned)(buf ^ 1) * (64u * 32u * 4u);
            const float* gA2 = gA + kk + 32;
            const float* gB2 = gB + kk + 32;
            async_ld_b128(abase + nb + eoff,      gA2);
            async_ld_b128(abase + nb + eoff + 16, gA2 + 4);
            async_ld_b128(bbase + nb + eoff,      gB2);
            async_ld_b128(bbase + nb + eoff + 16, gB2 + 4);
        }
        const float* Ap = &As[buf][0];
        const float* Bp = &Bs[buf][0];
        const int arow = wm * 16 + lrow;
        const int b0r  = wn * 32 + lrow;
        const int b1r  = b0r + 16;
#pragma unroll
        for (int ks = 0; ks < 32; ks += 4) {
            // ISA 7.12.2 f32 A/B frag: lanes 0-15 K={ks,ks+1}, lanes 16-31 K={ks+2,ks+3}
            v2f a  = *(const v2f*)(Ap + arow * 32 + ks + lhi * 2);
            v2f b0 = *(const v2f*)(Bp + b0r * 32 + ks + lhi * 2);
            v2f b1 = *(const v2f*)(Bp + b1r * 32 + ks + lhi * 2);
            acc0 = __builtin_amdgcn_wmma_f32_16x16x4_f32(false, a, false, b0, (short)0, acc0, false, false);
            acc1 = __builtin_amdgcn_wmma_f32_16x16x4_f32(false, a, false, b1, (short)0, acc1, false, false);
        }
        buf ^= 1;
    }

#pragma unroll
    for (int j = 0; j < 2; ++j) {
        v8f acc = j ? acc1 : acc0;
        int gn = n0 + wn * 32 + j * 16 + lrow;
        float bv = bias ? bias[gn] : 0.f;
#pragma unroll
        for (int rr = 0; rr < 8; ++rr) {
            int gm = m0 + wm * 16 + lhi * 8 + rr;   // C/D: m = rr + 8*(lane/16)
            long idx = (long)gm * N + gn;
            float v = acc[rr] * scale + bv;
            if (addsrc) v += addsrc[idx];
            if (relu) v = fmaxf(v, 0.f);
            C[idx] = v;
        }
    }
}

// ---------------------------------------------------------------------------
// out[c][r] = in[r][c]; rows%32==0, cols%32==0; per-batch stride rows*cols
__global__ void transpose_kernel(const float* __restrict__ in, float* __restrict__ out,
                                 int rows, int cols)
{
    __shared__ float tile[32][33];
    const int b = blockIdx.z;
    const float* ip = in + (long)b * rows * cols;
    float* op = out + (long)b * rows * cols;
    const int r0 = blockIdx.y * 32, c0 = blockIdx.x * 32;
    const int tx = threadIdx.x & 31, ty = threadIdx.x >> 5;   // 32 x 8
    for (int i = ty; i < 32; i += 8)
        tile[i][tx] = ip[(long)(r0 + i) * cols + c0 + tx];
    __syncthreads();
    for (int i = ty; i < 32; i += 8)
        op[(long)(c0 + i) * rows + r0 + tx] = tile[tx][i];
}

// feat[b,n,c] = w0[c,0:6] . corr[b,n,0:6] + b0[c]
__global__ void conv0_kernel(const float* __restrict__ corr, const float* __restrict__ w0,
                             const float* __restrict__ b0, float* __restrict__ feat)
{
    long i = (long)blockIdx.x * 256 + threadIdx.x;
    if (i >= (long)BS * N_PTS * CH) return;
    int c = (int)(i & (CH - 1));
    long p = i >> 7;
    const float* x = corr + p * 6;
    float a = b0[c];
#pragma unroll
    for (int j = 0; j < 6; ++j) a += w0[c * 6 + j] * x[j];
    feat[i] = a;
}

// compat[b,o,i] = max(1 - (|so-si| - |to-ti|)^2 / sigma_d^2, 0), ref pdist formula
__global__ void compat_kernel(const float* __restrict__ src, const float* __restrict__ tgt,
                              float* __restrict__ compat, int n)
{
    const int b = blockIdx.z;
    const int o = blockIdx.y * 16 + (threadIdx.x >> 4);
    const int i = blockIdx.x * 16 + (threadIdx.x & 15);
    const float* sp = src + (long)b * n * 3;
    const float* tp = tgt + (long)b * n * 3;
    float so0 = sp[o*3], so1 = sp[o*3+1], so2 = sp[o*3+2];
    float si0 = sp[i*3], si1 = sp[i*3+1], si2 = sp[i*3+2];
    float d2s = (so0*so0+so1*so1+so2*so2) + (si0*si0+si1*si1+si2*si2)
              - 2.f*(so0*si0+so1*si1+so2*si2);
    float ds = sqrtf(fmaxf(d2s, 0.f));
    float to0 = tp[o*3], to1 = tp[o*3+1], to2 = tp[o*3+2];
    float ti0 = tp[i*3], ti1 = tp[i*3+1], ti2 = tp[i*3+2];
    float d2t = (to0*to0+to1*to1+to2*to2) + (ti0*ti0+ti1*ti1+ti2*ti2)
              - 2.f*(to0*ti0+to1*ti1+to2*ti2);
    float dt = sqrtf(fmaxf(d2t, 0.f));
    float c = ds - dt;
    compat[((long)b * n + o) * n + i] = fmaxf(1.f - c * c * SIG_D2_INV, 0.f);
}

// BatchNorm stats over (b, n) per channel; x is [rows, ch] contiguous
__global__ void bn_stats_kernel(const float* __restrict__ x, float* __restrict__ mu,
                                float* __restrict__ rstd, int ch, int rows)
{
    const int c = blockIdx.x, t = threadIdx.x;
    __shared__ float red[256];
    float a = 0.f;
    for (int i = t; i < rows; i += 256) a += x[(long)i * ch + c];
    red[t] = a; __syncthreads();
    for (int off = 128; off > 0; off >>= 1) { if (t < off) red[t] += red[t + off]; __syncthreads(); }
    float m = red[0] / (float)rows;
    __syncthreads();
    float v = 0.f;
    for (int i = t; i < rows; i += 256) { float d = x[(long)i * ch + c] - m; v += d * d; }
    red[t] = v; __syncthreads();
    for (int off = 128; off > 0; off >>= 1) { if (t < off) red[t] += red[t + off]; __syncthreads(); }
    if (t == 0) { mu[c] = m; rstd[c] = rsqrtf(red[0] / (float)rows + BN_EPS); }
}

__global__ void bn_apply_kernel(const float* __restrict__ x, float* __restrict__ y,
                                const float* __restrict__ g, const float* __restrict__ beta,
                                const float* __restrict__ mu, const float* __restrict__ rstd,
                                long total, int ch, int relu)
{
    long i = (long)blockIdx.x * 256 + threadIdx.x;
    if (i >= total) return;
    int c = (int)(i % ch);
    float v = (x[i] - mu[c]) * rstd[c] * g[c] + beta[c];
    if (relu) v = fmaxf(v, 0.f);
    y[i] = v;
}

// In-place row softmax of (compat * att); one block per row
__global__ void softmax_kernel(float* __restrict__ att, const float* __restrict__ compat, int n)
{
    const int b = blockIdx.y, o = blockIdx.x, t = threadIdx.x;
    float* row = att + ((long)b * n + o) * n;
    const float* crow = compat + ((long)b * n + o) * n;
    __shared__ float red[256];
    float mx = -3.4e38f;
    for (int i = t; i < n; i += 256) mx = fmaxf(mx, crow[i] * row[i]);
    red[t] = mx; __syncthreads();
    for (int off = 128; off > 0; off >>= 1) { if (t < off) red[t] = fmaxf(red[t], red[t + off]); __syncthreads(); }
    mx = red[0]; __syncthreads();
    float sm = 0.f;
    for (int i = t; i < n; i += 256) { float e = expf(crow[i] * row[i] - mx); row[i] = e; sm += e; }
    red[t] = sm; __syncthreads();
    for (int off = 128; off > 0; off >>= 1) { if (t < off) red[t] += red[t + off]; __syncthreads(); }
    float inv = 1.f / red[0];
    for (int i = t; i < n; i += 256) row[i] *= inv;
}

// nf = feat / max(||feat_row||, 1e-12); block = 128 threads = CH
__global__ void normalize_kernel(const float* __restrict__ feat, float* __restrict__ nf, int n)
{
    const int b = blockIdx.y;
    const long row = (long)b * n + blockIdx.x;
    const int t = threadIdx.x;
    __shared__ float red[128];
    float v = feat[row * CH + t];
    red[t] = v * v; __syncthreads();
    for (int off = 64; off > 0; off >>= 1) { if (t < off) red[t] += red[t + off]; __syncthreads(); }
    float inv = 1.f / fmaxf(sqrtf(red[0]), 1e-12f);
    nf[row * CH + t] = v * inv;
}

// M = clip(1 - (1-S)/sigma^2, 0, 1) with zero diagonal
__global__ void finalize_M_kernel(const float* __restrict__ S, float* __restrict__ Mout,
                                  const float* __restrict__ sigma, int n)
{
    long i = (long)blockIdx.x * 256 + threadIdx.x;
    long nn = (long)n * n;
    if (i >= (long)BS * nn) return;
    long r = i % nn;
    int row = (int)(r / n), col = (int)(r % n);
    float sg = sigma[0];
    float m = 1.f - (1.f - S[i]) / (sg * sg);
    m = fminf(fmaxf(m, 0.f), 1.f);
    Mout[i] = (row == col) ? 0.f : m;
}

// confidence head: 128 -> 32 -> 32 -> 1 per point; block 128 threads (1 pt each)
__global__ __launch_bounds__(128) void conf_head_kernel(
    const float* __restrict__ feat,
    const float* __restrict__ cw1, const float* __restrict__ cb1,
    const float* __restrict__ cw2, const float* __restrict__ cb2,
    const float* __restrict__ cw3, const float* __restrict__ cb3,
    float* __restrict__ conf, int n)
{
    __shared__ float W1[32 * 128];
    __shared__ float W2[32 * 32];
    __shared__ float B1[32], B2[32], W3[32];
    __shared__ float B3;
    const int t = threadIdx.x;
    for (int i = t; i < 32 * 128; i += 128) W1[i] = cw1[i];
    for (int i = t; i < 32 * 32; i += 128) W2[i] = cw2[i];
    if (t < 32) { B1[t] = cb1[t]; B2[t] = cb2[t]; W3[t] = cw3[t]; }
    if (t == 0) B3 = cb3[0];
    __syncthreads();
    const int b = blockIdx.y;
    const int p = blockIdx.x * 128 + t;
    const float* x = feat + ((long)b * n + p) * CH;
    float h1[32], h2[32];
#pragma unroll 4
    for (int o = 0; o < 32; ++o) {
        float a = B1[o];
        for (int c = 0; c < 128; ++c) a += W1[o * 128 + c] * x[c];
        h1[o] = fmaxf(a, 0.f);
    }
#pragma unroll 4
    for (int o = 0; o < 32; ++o) {
        float a = B2[o];
        for (int c = 0; c < 32; ++c) a += W2[o * 32 + c] * h1[c];
        h2[o] = fmaxf(a, 0.f);
    }
    float out = B3;
#pragma unroll
    for (int c = 0; c < 32; ++c) out += W3[c] * h2[c];
    conf[(long)b * n + p] = out;
}

// iterative top-NSEED selection (ties -> lower index, matching lax.top_k)
__global__ void topk_seeds_kernel(const float* __restrict__ conf, int* __restrict__ seeds,
                                  int n, int ns)
{
    const int b = blockIdx.x, t = threadIdx.x;
    const float* c = conf + (long)b * n;
    __shared__ float rv[256];
    __shared__ int ri[256];
    __shared__ unsigned char used[N_PTS];
    for (int i = t; i < n; i += 256) used[i] = 0;
    __syncthreads();
    for (int s = 0; s < ns; ++s) {
        float bv = -3.4e38f; int bi = n;
        for (int i = t; i < n; i += 256)
            if (!used[i]) { float v = c[i]; if (v > bv) { bv = v; bi = i; } }
        rv[t] = bv; ri[t] = bi; __syncthreads();
        for (int off = 128; off > 0; off >>= 1) {
            if (t < off) {
                if (rv[t + off] > rv[t] || (rv[t + off] == rv[t] && ri[t + off] < ri[t])) {
                    rv[t] = rv[t + off]; ri[t] = ri[t + off];
                }
            }
            __syncthreads();
        }
        if (t == 0) { seeds[(long)b * ns + s] = ri[0]; used[ri[0]] = 1; }
        __syncthreads();
    }
}

// ---- 3x3 rigid R from H = U S V^T  ->  R = V diag(1,1,det(VU^T)) U^T -------
__device__ void rigid_R_from_H(const float H[9], float R[9])
{
    float a[3][3], vc[3][3];   // columns
    for (int j = 0; j < 3; ++j)
        for (int i = 0; i < 3; ++i) { a[j][i] = H[i * 3 + j]; vc[j][i] = (i == j) ? 1.f : 0.f; }
    for (int sweep = 0; sweep < 12; ++sweep) {
        for (int p = 0; p < 2; ++p) for (int q = p + 1; q < 3; ++q) {
            float app = 0.f, bqq = 0.f, apq = 0.f;
            for (int i = 0; i < 3; ++i) {
                app += a[p][i] * a[p][i]; bqq += a[q][i] * a[q][i]; apq += a[p][i] * a[q][i];
            }
            if (fabsf(apq) < 1e-18f) continue;
            float tau = (bqq - app) / (2.f * apq);
            float t2 = (tau >= 0.f ? 1.f : -1.f) / (fabsf(tau) + sqrtf(1.f + tau * tau));
            float cc = 1.f / sqrtf(1.f + t2 * t2);
            float ss = cc * t2;
            for (int i = 0; i < 3; ++i) {
                float ap = a[p][i], aq = a[q][i];
                a[p][i] = cc * ap - ss * aq; a[q][i] = ss * ap + cc * aq;
                float vp = vc[p][i], vq = vc[q][i];
                vc[p][i] = cc * vp - ss * vq; vc[q][i] = ss * vp + cc * vq;
            }
        }
    }
    float sv[3];
    for (int j = 0; j < 3; ++j) sv[j] = sqrtf(a[j][0]*a[j][0] + a[j][1]*a[j][1] + a[j][2]*a[j][2]);
    int idx[3] = {0, 1, 2};
    if (sv[idx[0]] < sv[idx[1]]) { int tt = idx[0]; idx[0] = idx[1]; idx[1] = tt; }
    if (sv[idx[0]] < sv[idx[2]]) { int tt = idx[0]; idx[0] = idx[2]; idx[2] = tt; }
    if (sv[idx[1]] < sv[idx[2]]) { int tt = idx[1]; idx[1] = idx[2]; idx[2] = tt; }
    float u[3][3], vv[3][3], s0 = sv[idx[0]], s1 = sv[idx[1]], s2 = sv[idx[2]];
    for (int j = 0; j < 3; ++j)
        for (int i = 0; i < 3; ++i) { u[j][i] = a[idx[j]][i]; vv[j][i] = vc[idx[j]][i]; }
    float i0 = 1.f / fmaxf(s0, 1e-12f), i1 = 1.f / fmaxf(s1, 1e-12f);
    for (int i = 0; i < 3; ++i) { u[0][i] *= i0; u[1][i] *= i1; }
    if (s2 > 1e-9f) { float i2 = 1.f / s2; for (int i = 0; i < 3; ++i) u[2][i] *= i2; }
    else {
        u[2][0] = u[0][1]*u[1][2] - u[0][2]*u[1][1];
        u[2][1] = u[0][2]*u[1][0] - u[0][0]*u[1][2];
        u[2][2] = u[0][0]*u[1][1] - u[0][1]*u[1][0];
    }
    float detU = u[0][0]*(u[1][1]*u[2][2]-u[1][2]*u[2][1])
               + u[0][1]*(u[1][2]*u[2][0]-u[1][0]*u[2][2])
               + u[0][2]*(u[1][0]*u[2][1]-u[1][1]*u[2][0]);
    float detV = vv[0][0]*(vv[1][1]*vv[2][2]-vv[1][2]*vv[2][1])
               + vv[0][1]*(vv[1][2]*vv[2][0]-vv[1][0]*vv[2][2])
               + vv[0][2]*(vv[1][0]*vv[2][1]-vv[1][1]*vv[2][0]);
    float e2 = detU * detV;
    for (int i = 0; i < 3; ++i)
        for (int j = 0; j < 3; ++j)
            R[i * 3 + j] = vv[0][i]*u[0][j] + vv[1][i]*u[1][j] + e2 * vv[2][i]*u[2][j];
}

// Per-seed: kNN from S row, 40x40 spectral matrix, power iteration, rigid fit, fitness
__global__ __launch_bounds__(128) void seed_solve_kernel(
    const float* __restrict__ S, const float* __restrict__ nf,
    const float* __restrict__ src, const float* __restrict__ tgt,
    const int* __restrict__ seeds, const float* __restrict__ sigma_p,
    float* __restrict__ trans, float* __restrict__ fitness, int n, int ns)
{
    const int b = blockIdx.y, si = blockIdx.x, t = threadIdx.x;
    __shared__ int knn[KNNK];
    __shared__ float kf[KNNK * CH];
    __shared__ float Ap[KNNK * 3], Bp[KNNK * 3];
    __shared__ float totM[KNNK * KNNK];
    __shared__ float vvb[KNNK], vtb[KNNK];
    __shared__ unsigned char used[N_PTS];
    __shared__ float rv[128];
    __shared__ int ri[128];
    __shared__ float RT[12];
    __shared__ float snorm;

    const int sd = seeds[(long)b * ns + si];
    const float* Srow = S + ((long)b * n + sd) * n;
    for (int i = t; i < n; i += 128) used[i] = 0;
    __syncthreads();
    for (int kth = 0; kth < KNNK; ++kth) {
        float bv = -3.4e38f; int bi = n;
        for (int i = t; i < n; i += 128)
            if (!used[i]) { float v = Srow[i]; if (v > bv) { bv = v; bi = i; } }
        rv[t] = bv; ri[t] = bi; __syncthreads();
        for (int off = 64; off > 0; off >>= 1) {
            if (t < off) {
                if (rv[t + off] > rv[t] || (rv[t + off] == rv[t] && ri[t + off] < ri[t])) {
                    rv[t] = rv[t + off]; ri[t] = ri[t + off];
                }
            }
            __syncthreads();
        }
        if (t == 0) { knn[kth] = ri[0]; used[ri[0]] = 1; }
        __syncthreads();
    }
    for (int i = t; i < KNNK * CH; i += 128)
        kf[i] = nf[((long)b * n + knn[i >> 7]) * CH + (i & (CH - 1))];
    for (int i = t; i < KNNK * 3; i += 128) {
        int r = i / 3, c = i - r * 3;
        Ap[i] = src[((long)b * n + knn[r]) * 3 + c];
        Bp[i] = tgt[((long)b * n + knn[r]) * 3 + c];
    }
    __syncthreads();
    const float sg = sigma_p[0];
    const float inv_s2 = 1.f / (sg * sg);
    for (int p = t; p < KNNK * KNNK; p += 128) {
        int i = p / KNNK, j = p - i * KNNK;
        float dot = 0.f;
        for (int c = 0; c < CH; ++c) dot += kf[i * CH + c] * kf[j * CH + c];
        float fm = fmaxf(1.f - (1.f - dot) * inv_s2, 0.f);
        float ds2 = 1e-12f, dt2 = 1e-12f;
        for (int c = 0; c < 3; ++c) {
            float d1 = Ap[i * 3 + c] - Ap[j * 3 + c]; ds2 += d1 * d1;
            float d2 = Bp[i * 3 + c] - Bp[j * 3 + c]; dt2 += d2 * d2;
        }
        float dd = sqrtf(ds2) - sqrtf(dt2);
        float sm = fmaxf(1.f - dd * dd * SIG_D2_INV, 0.f);
        totM[p] = (i == j) ? 0.f : fm * sm;
    }
    __syncthreads();
    if (t < KNNK) vvb[t] = 1.f;
    __syncthreads();
    for (int it = 0; it < 10; ++it) {
        if (t < KNNK) {
            float a = 0.f;
            for (int j = 0; j < KNNK; ++j) a += totM[t * KNNK + j] * vvb[j];
            vtb[t] = a;
        }
        __syncthreads();
        if (t == 0) {
            float a = 0.f;
            for (int j = 0; j < KNNK; ++j) a += vtb[j] * vtb[j];
            snorm = sqrtf(a) + 1e-6f;
        }
        __syncthreads();
        if (t < KNNK) vvb[t] = vtb[t] / snorm;
        __syncthreads();
    }
    if (t == 0) {
        float sum = 0.f;
        for (int j = 0; j < KNNK; ++j) sum += vvb[j];
        float inv = 1.f / (sum + 1e-6f);
        float w[KNNK]; float wsum = 0.f;
        for (int j = 0; j < KNNK; ++j) { w[j] = fmaxf(vvb[j] * inv, 0.f); wsum += w[j]; }
        float denom = wsum + 1e-6f;
        float cA[3] = {0, 0, 0}, cB[3] = {0, 0, 0};
        for (int j = 0; j < KNNK; ++j)
            for (int c = 0; c < 3; ++c) { cA[c] += Ap[j*3+c] * w[j]; cB[c] += Bp[j*3+c] * w[j]; }
        for (int c = 0; c < 3; ++c) { cA[c] /= denom; cB[c] /= denom; }
        float H[9] = {0, 0, 0, 0, 0, 0, 0, 0, 0};
        for (int j = 0; j < KNNK; ++j)
            for (int r = 0; r < 3; ++r)
                for (int c = 0; c < 3; ++c)
                    H[r*3+c] += (Ap[j*3+r] - cA[r]) * w[j] * (Bp[j*3+c] - cB[c]);
        float R[9];
        rigid_R_from_H(H, R);
        float tv[3];
        for (int r = 0; r < 3; ++r)
            tv[r] = cB[r] - (R[r*3]*cA[0] + R[r*3+1]*cA[1] + R[r*3+2]*cA[2]);
        float* T = trans + ((long)b * ns + si) * 16;
        for (int r = 0; r < 3; ++r) {
            T[r*4+0] = R[r*3+0]; T[r*4+1] = R[r*3+1]; T[r*4+2] = R[r*3+2]; T[r*4+3] = tv[r];
        }
        T[12] = 0.f; T[13] = 0.f; T[14] = 0.f; T[15] = 1.f;
        for (int r = 0; r < 9; ++r) RT[r] = R[r];
        RT[9] = tv[0]; RT[10] = tv[1]; RT[11] = tv[2];
    }
    __syncthreads();
    int cnt = 0;
    for (int i = t; i < n; i += 128) {
        const float* sp = src + ((long)b * n + i) * 3;
        const float* tp = tgt + ((long)b * n + i) * 3;
        float px = RT[0]*sp[0] + RT[1]*sp[1] + RT[2]*sp[2] + RT[9];
        float py = RT[3]*sp[0] + RT[4]*sp[1] + RT[5]*sp[2] + RT[10];
        float pz = RT[6]*sp[0] + RT[7]*sp[1] + RT[8]*sp[2] + RT[11];
        float dx = px - tp[0], dy = py - tp[1], dz = pz - tp[2];
        float L2 = sqrtf(dx*dx + dy*dy + dz*dz + 1e-12f);
        if (L2 < 0.1f) cnt++;
    }
    ri[t] = cnt; __syncthreads();
    for (int off = 64; off > 0; off >>= 1) { if (t < off) ri[t] += ri[t + off]; __syncthreads(); }
    if (t == 0) fitness[(long)b * ns + si] = (float)ri[0] / (float)n;
}

__global__ void final_select_kernel(const float* __restrict__ fitness,
                                    const float* __restrict__ trans,
                                    float* __restrict__ out, int ns)
{
    const int b = blockIdx.x;
    if (threadIdx.x != 0) return;
    int best = 0; float bv = fitness[(long)b * ns];
    for (int i = 1; i < ns; ++i) {
        float v = fitness[(long)b * ns + i];
        if (v > bv) { bv = v; best = i; }
    }
    for (int k = 0; k < 16; ++k) out[b * 16 + k] = trans[((long)b * ns + best) * 16 + k];
}

// ---------------------------------------------------------------------------
static inline void launch_gemm(hipStream_t s,
                               const float* A, const float* B, float* C,
                               const float* bias, const float* addsrc,
                               int M, int N, int K, long sA, long sB, long sC,
                               float scale, int relu)
{
    dim3 g(N / 64, M / 64, BS);
    gemm_wmma_f32<<<g, 256, 0, s>>>(A, B, C, bias, addsrc, M, N, K, sA, sB, sC, scale, relu);
}

extern "C" void kernel_launch(void* const* d_in, const int* in_sizes, int n_in,
                              void* d_out, int out_size, void* d_ws, size_t ws_size,
                              hipStream_t stream)
{
    if (n_in < 132) return;
    const float* corr = (const float*)d_in[0];
    const float* src  = (const float*)d_in[1];
    const float* tgt  = (const float*)d_in[2];
    // params flattened jax-style (sorted keys):
    // 3:b0 4:cb1 5:cb2 6:cb3 7:cw1 8:cw2 9:cw3 10..129: 6 layers x 20, 130:sigma 131:w0
    const float* b0p = (const float*)d_in[3];
    const float* cb1 = (const float*)d_in[4];
    const float* cb2 = (const float*)d_in[5];
    const float* cb3 = (const float*)d_in[6];
    const float* cw1 = (const float*)d_in[7];
    const float* cw2 = (const float*)d_in[8];
    const float* cw3 = (const float*)d_in[9];
    const float* sigma = (const float*)d_in[130];
    const float* w0p   = (const float*)d_in[131];
    // layer leaves sorted: b1 b2 b3 bc bk bq bt1 bt2 btc bv g1 g2 gc w1 w2 w3 wc wk wq wv
    enum { L_B1 = 0, L_B2, L_B3, L_BC, L_BK, L_BQ, L_BT1, L_BT2, L_BTC, L_BV,
           L_G1, L_G2, L_GC, L_W1, L_W2, L_W3, L_WC, L_WK, L_WQ, L_WV };
    auto LP = [&](int l, int k) -> const float* { return (const float*)d_in[10 + l * 20 + k]; };

    // workspace layout
    size_t off = 0;
    char* wsb = (char*)d_ws;
    auto alloc = [&](size_t bytes) -> void* {
        void* p = wsb + off;
        off += (bytes + 255) & ~(size_t)255;
        return p;
    };
    const long sF  = (long)N_PTS * CH;           // per-batch feature stride
    const long sH  = (long)N_PTS * 64;           // 64-channel stride
    const long sNN = (long)N_PTS * N_PTS;        // n x n stride
    float* feat   = (float*)alloc((size_t)BS * sF * 4);
    float* tmpA   = (float*)alloc((size_t)BS * sF * 4);
    float* tmpB   = (float*)alloc((size_t)BS * sF * 4);
    float* qb     = (float*)alloc((size_t)BS * sF * 4);
    float* kb     = (float*)alloc((size_t)BS * sF * 4);
    float* vb     = (float*)alloc((size_t)BS * sF * 4);
    float* vtb2   = (float*)alloc((size_t)BS * sF * 4);    // V^T per layer
    float* nfb    = (float*)alloc((size_t)BS * sF * 4);
    float* attb   = (float*)alloc((size_t)BS * sNN * 4);   // att -> softmax w -> S
    float* compat = (float*)alloc((size_t)BS * sNN * 4);
    float* bn_mu  = (float*)alloc(CH * 4);
    float* bn_rs  = (float*)alloc(CH * 4);
    int*   seeds  = (int*)alloc((size_t)BS * NSEED * 4);
    float* transb = (float*)alloc((size_t)BS * NSEED * 16 * 4);
    float* fitb   = (float*)alloc((size_t)BS * NSEED * 4);
    if (off > ws_size) return;

    float* out_trans = (float*)d_out;
    float* out_conf  = out_trans + BS * 16;
    float* out_M     = out_conf + (long)BS * N_PTS;

    // 1) input embedding + compat matrix
    conv0_kernel<<<(BS * N_PTS * CH) / 256, 256, 0, stream>>>(corr, w0p, b0p, feat);
    compat_kernel<<<dim3(N_PTS / 16, N_PTS / 16, BS), 256, 0, stream>>>(src, tgt, compat, N_PTS);

    const float att_scale = 0.08838834764831845f;  // 1/sqrt(128)
    for (int l = 0; l < 6; ++l) {
        // feat = relu(bn(conv(feat)))
        launch_gemm(stream, feat, LP(l, L_WC), tmpA, LP(l, L_BC), nullptr,
                    N_PTS, CH, CH, sF, 0, sF, 1.f, 0);
        bn_stats_kernel<<<CH, 256, 0, stream>>>(tmpA, bn_mu, bn_rs, CH, BS * N_PTS);
        bn_apply_kernel<<<(BS * N_PTS * CH) / 256, 256, 0, stream>>>(
            tmpA, feat, LP(l, L_GC), LP(l, L_BTC), bn_mu, bn_rs, (long)BS * N_PTS * CH, CH, 1);
        // Q, K, V
        launch_gemm(stream, feat, LP(l, L_WQ), qb, LP(l, L_BQ), nullptr,
                    N_PTS, CH, CH, sF, 0, sF, 1.f, 0);
        launch_gemm(stream, feat, LP(l, L_WK), kb, LP(l, L_BK), nullptr,
                    N_PTS, CH, CH, sF, 0, sF, 1.f, 0);
        launch_gemm(stream, feat, LP(l, L_WV), vb, LP(l, L_BV), nullptr,
                    N_PTS, CH, CH, sF, 0, sF, 1.f, 0);
        // att = Q K^T / sqrt(C); w = softmax(compat * att) (in-place)
        launch_gemm(stream, qb, kb, attb, nullptr, nullptr,
                    N_PTS, N_PTS, CH, sF, sF, sNN, att_scale, 0);
        softmax_kernel<<<dim3(N_PTS, BS), 256, 0, stream>>>(attb, compat, N_PTS);
        // msg = w V  (as NT gemm against V^T)
        transpose_kernel<<<dim3(CH / 32, N_PTS / 32, BS), 256, 0, stream>>>(vb, vtb2, N_PTS, CH);
        launch_gemm(stream, attb, vtb2, tmpA, nullptr, nullptr,
                    N_PTS, CH, N_PTS, sNN, sF, sF, 1.f, 0);
        // bottleneck: 128 -> 64 -> 64 -> 128 (+ residual)
        launch_gemm(stream, tmpA, LP(l, L_W1), tmpB, LP(l, L_B1), nullptr,
                    N_PTS, 64, CH, sF, 0, sH, 1.f, 0);
        bn_stats_kernel<<<64, 256, 0, stream>>>(tmpB, bn_mu, bn_rs, 64, BS * N_PTS);
        bn_apply_kernel<<<(BS * N_PTS * 64) / 256, 256, 0, stream>>>(
            tmpB, tmpB, LP(l, L_G1), LP(l, L_BT1), bn_mu, bn_rs, (long)BS * N_PTS * 64, 64, 1);
        launch_gemm(stream, tmpB, LP(l, L_W2), tmpA, LP(l, L_B2), nullptr,
                    N_PTS, 64, 64, sH, 0, sH, 1.f, 0);
        bn_stats_kernel<<<64, 256, 0, stream>>>(tmpA, bn_mu, bn_rs, 64, BS * N_PTS);
        bn_apply_kernel<<<(BS * N_PTS * 64) / 256, 256, 0, stream>>>(
            tmpA, tmpA, LP(l, L_G2), LP(l, L_BT2), bn_mu, bn_rs, (long)BS * N_PTS * 64, 64, 1);
        launch_gemm(stream, tmpA, LP(l, L_W3), feat, LP(l, L_B3), feat,
                    N_PTS, CH, 64, sH, 0, sF, 1.f, 0);
    }

    // 2) normalized features, S = nf nf^T, M output
    normalize_kernel<<<dim3(N_PTS, BS), 128, 0, stream>>>(feat, nfb, N_PTS);
    launch_gemm(stream, nfb, nfb, attb, nullptr, nullptr,
                N_PTS, N_PTS, CH, sF, sF, sNN, 1.f, 0);
    finalize_M_kernel<<<(int)((BS * sNN) / 256), 256, 0, stream>>>(attb, out_M, sigma, N_PTS);

    // 3) confidence head + seeds
    conf_head_kernel<<<dim3(N_PTS / 128, BS), 128, 0, stream>>>(
        feat, cw1, cb1, cw2, cb2, cw3, cb3, out_conf, N_PTS);
    topk_seeds_kernel<<<BS, 256, 0, stream>>>(out_conf, seeds, N_PTS, NSEED);

    // 4) per-seed spectral matching + rigid fit + fitness, final selection
    seed_solve_kernel<<<dim3(NSEED, BS), 128, 0, stream>>>(
        attb, nfb, src, tgt, seeds, sigma, transb, fitb, N_PTS, NSEED);
    final_select_kernel<<<BS, 32, 0, stream>>>(fitb, transb, out_trans, NSEED);
}